// LUTTreeKernel_78743930405436
// MI455X (gfx1250) — compile-verified
//
#include <hip/hip_runtime.h>

// ---------------------------------------------------------------------------
// Differentiable LUT-tree forward for MI455X (gfx1250).
//
// Roofline: 128 MB streamed from HBM (~5.5 us @ 23.3 TB/s) vs ~0.5 GFLOP of
// lerps (~3.75 flop/byte) -> near the HBM/VALU balance point. Not a GEMM
// (corner weights are data-dependent per node), so WMMA would waste >=14/16
// of its throughput on a block-diagonal expansion. The CDNA5 path that fits:
//   * GLOBAL_LOAD_ASYNC_TO_LDS_B128 streams x tiles into LDS, triple-
//     buffered (2 chunks in flight), synchronized with s_wait_asynccnt.
//   * All LUT tables (21.8 KB) are staged into LDS once per block, so the
//     inner loop reads them as uniform-address ds_load broadcasts instead of
//     per-lane VMEM loads.
//   * Lerp-tree evaluation: 15 v_sub + 15 v_fma per node, deep ILP.
// ---------------------------------------------------------------------------

#define BLOCK   64              // threads per block (2 wave32)
#define ROWS    64              // batch rows per block (1 row per thread)
#define KCHUNK  64              // k-floats per chunk = one layer-2 subtree
#define NCHUNK  16              // 1024 / 64
#define LROW    68              // padded floats per LDS row (272B, 16B aligned)
#define TILEFL  (ROWS * LROW)   // floats per LDS tile buffer
#define NBUF    3               // triple buffer, 2 chunks in flight
#define TBL0FL  4096            // t0: 256*16
#define TBL1FL  1024            // t1:  64*16
#define TBL2FL  256             // t2:  16*16
#define TBL3FL  64              // t3:   4*16
#define TBL4FL  16              // t4:   1*16
#define TBLFL   (TBL0FL + TBL1FL + TBL2FL + TBL3FL + TBL4FL)  // 5456 floats

// 4-input multilinear interpolation over a 16-entry table, evaluated as a
// lerp tree (matches reference corner ordering: bit i of corner c selects
// x_i vs 1-x_i, reduced LSB-first).
__device__ __forceinline__ float lut4(const float* __restrict__ tb,
                                      float x0, float x1, float x2, float x3) {
  float a0 = fmaf(x0, tb[1]  - tb[0],  tb[0]);
  float a1 = fmaf(x0, tb[3]  - tb[2],  tb[2]);
  float a2 = fmaf(x0, tb[5]  - tb[4],  tb[4]);
  float a3 = fmaf(x0, tb[7]  - tb[6],  tb[6]);
  float a4 = fmaf(x0, tb[9]  - tb[8],  tb[8]);
  float a5 = fmaf(x0, tb[11] - tb[10], tb[10]);
  float a6 = fmaf(x0, tb[13] - tb[12], tb[12]);
  float a7 = fmaf(x0, tb[15] - tb[14], tb[14]);
  float b0 = fmaf(x1, a1 - a0, a0);
  float b1 = fmaf(x1, a3 - a2, a2);
  float b2 = fmaf(x1, a5 - a4, a4);
  float b3 = fmaf(x1, a7 - a6, a6);
  float c0 = fmaf(x2, b1 - b0, b0);
  float c1 = fmaf(x2, b3 - b2, b2);
  return fmaf(x3, c1 - c0, c0);
}

// Issue one chunk of the x-tile as 16 per-lane async 16B copies to LDS.
// Lane layout: r0 = tid>>4, c4 = tid&15; a wave covers two contiguous
// 256B row segments per instruction -> fully coalesced.
// Ordering vs. consumers is enforced by s_wait_asynccnt + __syncthreads().
__device__ __forceinline__ void issue_chunk(const float* gbase,
                                            unsigned lbase_bytes,
                                            int kc, int buf) {
  const float* g = gbase + kc * KCHUNK;
  const unsigned l0 = lbase_bytes + (unsigned)buf * (TILEFL * 4u);
#pragma unroll
  for (int i = 0; i < 16; ++i) {
    const unsigned l = l0 + (unsigned)i * (4u * LROW * 4u); // +4 rows per step
    const float* gp = g + (size_t)i * 4 * 1024;             // +4 batch rows
    asm volatile("global_load_async_to_lds_b128 %0, %1, off"
                 :: "v"(l), "v"(gp));
  }
}

// Copy a table from global memory into LDS (block-cooperative, float4s).
__device__ __forceinline__ void stage_table(float* __restrict__ dst,
                                            const float* __restrict__ src,
                                            int nfloats, int tid) {
  for (int i = tid * 4; i < nfloats; i += BLOCK * 4) {
    *reinterpret_cast<float4*>(dst + i) =
        *reinterpret_cast<const float4*>(src + i);
  }
}

__global__ __launch_bounds__(BLOCK)
__attribute__((amdgpu_waves_per_eu(4)))
void lut_tree_fwd(const float* __restrict__ x,
                  const float* __restrict__ t0,
                  const float* __restrict__ t1,
                  const float* __restrict__ t2,
                  const float* __restrict__ t3,
                  const float* __restrict__ t4,
                  float* __restrict__ out) {
  __shared__ float lds[NBUF * TILEFL + TBLFL];

  const int tid = threadIdx.x;
  const int b0  = blockIdx.x * ROWS;
  const int r0  = tid >> 4;
  const int c4  = tid & 15;

  // Low 32 bits of a generic pointer to LDS == wave-relative LDS byte offset.
  const unsigned lds0  = (unsigned)(unsigned long long)(&lds[0]);
  const unsigned lbase = lds0 + (unsigned)(r0 * LROW + c4 * 4) * 4u;
  const float*   gbase = x + (size_t)(b0 + r0) * 1024 + (c4 * 4);

  // Prologue: two x-chunks in flight via the async data mover.
  issue_chunk(gbase, lbase, 0, 0);
  issue_chunk(gbase, lbase, 1, 1);

  // Stage LUT tables into LDS (L2-resident source; once per block).
  float* lt0 = &lds[NBUF * TILEFL];
  float* lt1 = lt0 + TBL0FL;
  float* lt2 = lt1 + TBL1FL;
  float* lt3 = lt2 + TBL2FL;
  float* lt4 = lt3 + TBL3FL;
  stage_table(lt0, t0, TBL0FL, tid);
  stage_table(lt1, t1, TBL1FL, tid);
  stage_table(lt2, t2, TBL2FL, tid);
  stage_table(lt3, t3, TBL3FL, tid);
  stage_table(lt4, t4, TBL4FL, tid);

  float n3v0 = 0.f, n3v1 = 0.f, n3v2 = 0.f, n3v3 = 0.f;

  for (int kg = 0; kg < 4; ++kg) {        // rolled: layer-3 groups
    float c2v0 = 0.f, c2v1 = 0.f, c2v2 = 0.f, c2v3 = 0.f;
#pragma unroll
    for (int k4 = 0; k4 < 4; ++k4) {      // unrolled: 4 chunks (layer-2 nodes)
      const int kc  = kg * 4 + k4;
      const int buf = kc % NBUF;

      if (kc + 2 < NCHUNK)
        issue_chunk(gbase, lbase, kc + 2, (kc + 2) % NBUF);

      // In-order per-wave ASYNCcnt: <=32 outstanding => chunk kc complete.
      if (kc < NCHUNK - 2)       asm volatile("s_wait_asynccnt 0x20" ::: "memory");
      else if (kc == NCHUNK - 2) asm volatile("s_wait_asynccnt 0x10" ::: "memory");
      else                       asm volatile("s_wait_asynccnt 0x0"  ::: "memory");
      __syncthreads();           // all waves' async writes (and staged tables)
                                 // now visible

      const float* row = &lds[buf * TILEFL + tid * LROW];

      float n1v[4];
#pragma unroll
      for (int j = 0; j < 4; ++j) {       // layer-1 nodes of this chunk
        float n0v[4];
#pragma unroll
        for (int ii = 0; ii < 4; ++ii) {  // layer-0 nodes
          const int i = j * 4 + ii;
          const float4 xv = *reinterpret_cast<const float4*>(row + i * 4);
          n0v[ii] = lut4(lt0 + (kc * 16 + i) * 16, xv.x, xv.y, xv.z, xv.w);
        }
        n1v[j] = lut4(lt1 + (kc * 4 + j) * 16, n0v[0], n0v[1], n0v[2], n0v[3]);
      }
      const float c2 = lut4(lt2 + kc * 16, n1v[0], n1v[1], n1v[2], n1v[3]);
      if      (k4 == 0) c2v0 = c2;
      else if (k4 == 1) c2v1 = c2;
      else if (k4 == 2) c2v2 = c2;
      else              c2v3 = c2;

      __syncthreads();           // reads done before buffer is overwritten
    }
    const float v3 = lut4(lt3 + kg * 16, c2v0, c2v1, c2v2, c2v3);
    if      (kg == 0) n3v0 = v3;
    else if (kg == 1) n3v1 = v3;
    else if (kg == 2) n3v2 = v3;
    else              n3v3 = v3;
  }

  out[b0 + tid] = lut4(lt4, n3v0, n3v1, n3v2, n3v3);
}

extern "C" void kernel_launch(void* const* d_in, const int* in_sizes, int n_in,
                              void* d_out, int out_size, void* d_ws, size_t ws_size,
                              hipStream_t stream) {
  (void)n_in; (void)out_size; (void)d_ws; (void)ws_size;
  const float* x  = (const float*)d_in[0];
  const float* t0 = (const float*)d_in[1];
  const float* t1 = (const float*)d_in[2];
  const float* t2 = (const float*)d_in[3];
  const float* t3 = (const float*)d_in[4];
  const float* t4 = (const float*)d_in[5];
  float* out = (float*)d_out;

  const int batch = in_sizes[0] / 1024;   // 32768
  dim3 grid(batch / ROWS), block(BLOCK);
  hipLaunchKernelGGL(lut_tree_fwd, grid, block, 0, stream,
                     x, t0, t1, t2, t3, t4, out);
}